// GATConv_25778393711285
// MI455X (gfx1250) — compile-verified
//
#include <hip/hip_runtime.h>

#define N_NODES 100000
#define IN_F    128
#define H_HEADS 4
#define D_OUT   16
#define HD      64          // H*D
#define E_EDGES 3200000
#define SLOPE   0.2f

typedef __attribute__((ext_vector_type(2))) float v2f;
typedef __attribute__((ext_vector_type(4))) float v4f;
typedef __attribute__((ext_vector_type(8))) float v8f;

// ---------------- order-preserving float <-> uint map (no NaNs expected) ----
__device__ __forceinline__ unsigned flt_flip(float f) {
    unsigned u = __float_as_uint(f);
    return u ^ (((int)u >> 31) | 0x80000000u);
}
__device__ __forceinline__ float flt_unflip(unsigned u) {
    unsigned b = (u & 0x80000000u) ? (u ^ 0x80000000u) : ~u;
    return __uint_as_float(b);
}

// ---------------- K1: ft = feat @ W^T via V_WMMA_F32_16X16X4_F32 ------------
// block = 256 threads = 8 waves; wave (rt, ct): 16-row x 16-col tile.
// ct == head index (D == 16).  grid.x = N/32.
__global__ __launch_bounds__(256) void gat_gemm(const float* __restrict__ feat,
                                                const float* __restrict__ W,
                                                float* __restrict__ ft) {
    const int wave = threadIdx.x >> 5;
    const int lane = threadIdx.x & 31;
    const int rt   = wave >> 2;        // 0..1
    const int ct   = wave & 3;         // head 0..3
    const int r0   = blockIdx.x * 32 + rt * 16;
    const int c0   = ct * 16;
    const int m    = lane & 15;
    const int hi   = lane >> 4;        // lanes 16-31 hold K+2,K+3

    const float* arow = feat + (size_t)(r0 + m) * IN_F + (hi << 1);
    const float* brow = W    + (size_t)(c0 + m) * IN_F + (hi << 1);

    v8f acc = {};
#pragma unroll
    for (int k0 = 0; k0 < IN_F; k0 += 4) {
        v2f a = *(const v2f*)(arow + k0);   // A[M=r0+m][K=k0+2*hi .. +1]
        v2f b = *(const v2f*)(brow + k0);   // B[K][N=c0+m] = W[c0+m][K]
        acc = __builtin_amdgcn_wmma_f32_16x16x4_f32(
            false, a, false, b, (short)0, acc, false, false);
    }
    // C/D layout: VGPR r -> row (r + 8*hi), col m
    float* outp = ft + (size_t)(r0 + hi * 8) * HD + c0 + m;
#pragma unroll
    for (int r = 0; r < 8; ++r) outp[(size_t)r * HD] = acc[r];
}

// ---------------- K2: el/er = <ft[n,h,:], attn_{l,r}[h,:]> ------------------
__global__ __launch_bounds__(256) void gat_attn(const float* __restrict__ ft,
                                                const float* __restrict__ al,
                                                const float* __restrict__ ar,
                                                float* __restrict__ el,
                                                float* __restrict__ er) {
    int idx = blockIdx.x * blockDim.x + threadIdx.x;  // idx = n*H + h
    if (idx >= N_NODES * H_HEADS) return;
    const int h = idx & 3;
    const float* f = ft + (size_t)idx * D_OUT;        // (n*4+h)*16
    float sl = 0.f, sr = 0.f;
#pragma unroll
    for (int d = 0; d < D_OUT; ++d) {
        float v = f[d];
        sl += v * al[h * D_OUT + d];
        sr += v * ar[h * D_OUT + d];
    }
    el[idx] = sl;
    er[idx] = sr;
}

// ---------------- K_init: zero m/s/out every call ---------------------------
__global__ __launch_bounds__(256) void gat_init(unsigned* __restrict__ m_u,
                                                float* __restrict__ s,
                                                float* __restrict__ out) {
    int idx = blockIdx.x * blockDim.x + threadIdx.x;
    if (idx < N_NODES * H_HEADS) { m_u[idx] = 0u; s[idx] = 0.f; }
    if (idx < N_NODES * HD)      out[idx] = 0.f;
}

__device__ __forceinline__ float edge_score(float l, float r) {
    float e = l + r;
    return e >= 0.f ? e : SLOPE * e;
}

// ---------------- K3: per-dst segment max via atomicMax(u32) ----------------
__global__ __launch_bounds__(256) void gat_edge_max(const int* __restrict__ src,
                                                    const int* __restrict__ dst,
                                                    const float* __restrict__ el,
                                                    const float* __restrict__ er,
                                                    unsigned* __restrict__ m_u) {
    int i = blockIdx.x * blockDim.x + threadIdx.x;
    if (i >= E_EDGES) return;
    const int s_ = src[i], d_ = dst[i];
    v4f l = *(const v4f*)(el + (size_t)s_ * 4);
    v4f r = *(const v4f*)(er + (size_t)d_ * 4);
#pragma unroll
    for (int h = 0; h < 4; ++h)
        atomicMax(&m_u[d_ * 4 + h], flt_flip(edge_score(l[h], r[h])));
}

// ---------------- K4: segment sum of exp(e - m[dst]) ------------------------
__global__ __launch_bounds__(256) void gat_edge_sum(const int* __restrict__ src,
                                                    const int* __restrict__ dst,
                                                    const float* __restrict__ el,
                                                    const float* __restrict__ er,
                                                    const unsigned* __restrict__ m_u,
                                                    float* __restrict__ ssum) {
    int i = blockIdx.x * blockDim.x + threadIdx.x;
    if (i >= E_EDGES) return;
    const int s_ = src[i], d_ = dst[i];
    v4f l = *(const v4f*)(el + (size_t)s_ * 4);
    v4f r = *(const v4f*)(er + (size_t)d_ * 4);
#pragma unroll
    for (int h = 0; h < 4; ++h) {
        float e = edge_score(l[h], r[h]);
        float a = expf(e - flt_unflip(m_u[d_ * 4 + h]));
        atomicAdd(&ssum[d_ * 4 + h], a);
    }
}

// ---------------- K5: out[dst,h,:] += ft[src,h,:] * a/s ---------------------
// one thread per (edge, head): 16 scatter-atomics each.
__global__ __launch_bounds__(256) void gat_edge_agg(const int* __restrict__ src,
                                                    const int* __restrict__ dst,
                                                    const float* __restrict__ el,
                                                    const float* __restrict__ er,
                                                    const unsigned* __restrict__ m_u,
                                                    const float* __restrict__ ssum,
                                                    const float* __restrict__ ft,
                                                    float* __restrict__ out) {
    int t = blockIdx.x * blockDim.x + threadIdx.x;
    if (t >= E_EDGES * H_HEADS) return;
    const int i = t >> 2;
    const int h = t & 3;
    const int s_ = src[i], d_ = dst[i];
    float e = edge_score(el[s_ * 4 + h], er[d_ * 4 + h]);
    float w = expf(e - flt_unflip(m_u[d_ * 4 + h])) / ssum[d_ * 4 + h];
    const float* f = ft + (size_t)s_ * HD + h * D_OUT;
    float* o       = out + (size_t)d_ * HD + h * D_OUT;
#pragma unroll
    for (int d = 0; d < D_OUT; ++d)
        atomicAdd(o + d, f[d] * w);
}

// ---------------------------------------------------------------------------
extern "C" void kernel_launch(void* const* d_in, const int* in_sizes, int n_in,
                              void* d_out, int out_size, void* d_ws, size_t ws_size,
                              hipStream_t stream) {
    const float* feat = (const float*)d_in[0];
    const float* W    = (const float*)d_in[1];
    const float* al   = (const float*)d_in[2];
    const float* ar   = (const float*)d_in[3];
    const int*   src  = (const int*)d_in[4];
    const int*   dst  = (const int*)d_in[5];
    float* out = (float*)d_out;

    // workspace layout (floats): ft | el | er | m_u | s   (32 MB total)
    float*    ft  = (float*)d_ws;
    float*    el  = ft + (size_t)N_NODES * HD;
    float*    er  = el + (size_t)N_NODES * H_HEADS;
    unsigned* m_u = (unsigned*)(er + (size_t)N_NODES * H_HEADS);
    float*    s   = (float*)(m_u + (size_t)N_NODES * H_HEADS);

    const int nb_gemm = N_NODES / 32;                         // 3125
    const int nb_attn = (N_NODES * H_HEADS + 255) / 256;      // 1563
    const int nb_init = (N_NODES * HD + 255) / 256;           // 25000
    const int nb_edge = (E_EDGES + 255) / 256;                // 12500
    const int nb_agg  = (E_EDGES * H_HEADS + 255) / 256;      // 50000

    gat_gemm<<<nb_gemm, 256, 0, stream>>>(feat, W, ft);
    gat_attn<<<nb_attn, 256, 0, stream>>>(ft, al, ar, el, er);
    gat_init<<<nb_init, 256, 0, stream>>>(m_u, s, out);
    gat_edge_max<<<nb_edge, 256, 0, stream>>>(src, dst, el, er, m_u);
    gat_edge_sum<<<nb_edge, 256, 0, stream>>>(src, dst, el, er, m_u, s);
    gat_edge_agg<<<nb_agg, 256, 0, stream>>>(src, dst, el, er, m_u, s, ft, out);
}